// GAT_17377437680279
// MI455X (gfx1250) — compile-verified
//
#include <hip/hip_runtime.h>
#include <math.h>

#define NN      30000
#define EE      400000
#define IN_DIM  128
#define HEADS   4
#define DD      64
#define CCLS    40
#define BN_EPS  1e-5f
#define SLOPE   0.2f
#define MT      5        // M tiles per wave: 30000 = 16 * MT * 375

typedef __attribute__((ext_vector_type(2))) float v2f;
typedef __attribute__((ext_vector_type(8))) float v8f;

// ---------------------------------------------------------------------------
// GEMM: C[M x Nout] = A[M x K] * B[K x Nout], all row-major f32.
// V_WMMA_F32_16X16X4_F32 (f32 in / f32 acc -> exact precision vs reference).
// Each wave owns MT vertical 16x16 tiles (80 rows x 16 cols) and reuses the
// B fragment across all MT tiles: 7 loads per 5 WMMAs instead of 15.
// 8 waves per block span 128 output columns.
// A 16x4 frag: lanes 0-15 rows M=lane, K pair {0,1}; lanes 16-31 K pair {2,3}.
// B 4x16 frag: lane group selects K half, lane%16 = N column.
// C/D: lanes 0-15 rows 0-7 (vgpr 0-7), lanes 16-31 rows 8-15.
// ---------------------------------------------------------------------------
__global__ __launch_bounds__(256) void gemm_wmma_f32(
    const float* __restrict__ A, const float* __restrict__ B,
    float* __restrict__ C, int K, int Nout) {
  const int wave = threadIdx.x >> 5;
  const int lane = threadIdx.x & 31;
  const int half = lane >> 4;          // 0: K {0,1}, 1: K {2,3}
  const int l16  = lane & 15;
  const int m0 = blockIdx.x * (16 * MT);
  const int n0 = (blockIdx.y * 8 + wave) << 4;
  if (n0 >= Nout) return;              // wave-uniform: EXEC stays all-ones
  const int ncol = n0 + l16;
  const float* Ap = A + (size_t)(m0 + l16) * K + (half << 1);
  const size_t tstride = (size_t)16 * K;       // A stride between M tiles
  v8f acc[MT];
#pragma unroll
  for (int t = 0; t < MT; ++t) acc[t] = (v8f){};
  for (int k = 0; k < K; k += 16) {
#pragma unroll
    for (int kk = 0; kk < 16; kk += 4) {
      const int kb = k + kk;
      const int kr = kb + (half << 1);
      v2f b;
      b[0] = B[(size_t)kr * Nout + ncol];
      b[1] = B[(size_t)(kr + 1) * Nout + ncol];
#pragma unroll
      for (int t = 0; t < MT; ++t) {
        float2 av = *(const float2*)(Ap + tstride * t + kb);
        v2f a; a[0] = av.x; a[1] = av.y;
        acc[t] = __builtin_amdgcn_wmma_f32_16x16x4_f32(
            false, a, false, b, (short)0, acc[t], false, false);
      }
    }
    if (k + 16 < K)  // warm L1/L2 for the next K-block of B (global_prefetch_b8)
      __builtin_prefetch(&B[(size_t)(k + 16 + (half << 1)) * Nout + ncol], 0, 1);
  }
#pragma unroll
  for (int t = 0; t < MT; ++t) {
    float* Cp = C + (size_t)(m0 + t * 16 + (half << 3)) * Nout + ncol;
#pragma unroll
    for (int r = 0; r < 8; ++r) Cp[(size_t)r * Nout] = acc[t][r];
  }
}

// ---------------------------------------------------------------------------
__global__ void fill_f32(float* __restrict__ p, float v, long n) {
  long i = (long)blockIdx.x * blockDim.x + threadIdx.x;
  if (i < n) p[i] = v;
}

__device__ __forceinline__ void atomicMaxFloat(float* addr, float v) {
  // init to -inf; works across mixed signs with the int/uint trick
  if (v >= 0.f) atomicMax((int*)addr, __float_as_int(v));
  else          atomicMin((unsigned int*)addr, __float_as_uint(v));
}

// Pass A: score_e,h = attn_h . leaky_relu(fs[src] + fd[dst]); seg-max into m.
__global__ void edge_score_max(const float* __restrict__ fs,
                               const float* __restrict__ fd,
                               const int* __restrict__ src,
                               const int* __restrict__ dst,
                               const float* __restrict__ attn,
                               float* __restrict__ score,
                               float* __restrict__ m, int Dh) {
  int idx = blockIdx.x * blockDim.x + threadIdx.x;   // e*HEADS + h
  if (idx >= EE * HEADS) return;
  const int e = idx >> 2, h = idx & 3;
  const int s = src[e], d = dst[e];
  const int HD = HEADS * Dh;
  const float4* fsp = (const float4*)(fs + (size_t)s * HD + h * Dh);
  const float4* fdp = (const float4*)(fd + (size_t)d * HD + h * Dh);
  const float4* ap  = (const float4*)(attn + h * Dh);
  float acc = 0.f;
#pragma unroll 4
  for (int i = 0; i < Dh / 4; ++i) {
    float4 a = fsp[i], b = fdp[i], w = ap[i];
    float v;
    v = a.x + b.x; v = v > 0.f ? v : SLOPE * v; acc += w.x * v;
    v = a.y + b.y; v = v > 0.f ? v : SLOPE * v; acc += w.y * v;
    v = a.z + b.z; v = v > 0.f ? v : SLOPE * v; acc += w.z * v;
    v = a.w + b.w; v = v > 0.f ? v : SLOPE * v; acc += w.w * v;
  }
  score[idx] = acc;
  atomicMaxFloat(&m[(size_t)d * HEADS + h], acc);
}

// Pass B: ex = exp(score - m[dst]); denom[dst] += ex (score overwritten by ex)
__global__ void edge_exp_denom(float* __restrict__ score,
                               const float* __restrict__ m,
                               const int* __restrict__ dst,
                               float* __restrict__ denom) {
  int idx = blockIdx.x * blockDim.x + threadIdx.x;
  if (idx >= EE * HEADS) return;
  const int e = idx >> 2, h = idx & 3;
  const int d = dst[e];
  float ex = expf(score[idx] - m[(size_t)d * HEADS + h]);
  score[idx] = ex;
  atomicAdd(&denom[(size_t)d * HEADS + h], ex);
}

// Pass C: rst[dst] += alpha * fs[src]; one thread per (edge, 4 features).
__global__ void edge_aggregate(const float* __restrict__ ex,
                               const float* __restrict__ denom,
                               const float* __restrict__ fs,
                               const int* __restrict__ src,
                               const int* __restrict__ dst,
                               float* __restrict__ rst, int Dh) {
  const int HD = HEADS * Dh;
  const int g4 = HD >> 2;                 // float4 groups per edge
  long idx = (long)blockIdx.x * blockDim.x + threadIdx.x;
  if (idx >= (long)EE * g4) return;
  const int e  = (int)(idx / g4);
  const int f  = (int)(idx % g4) << 2;
  const int h  = f / Dh;                  // Dh % 4 == 0, no straddle
  const int s = src[e], d = dst[e];
  const float den = denom[(size_t)d * HEADS + h];
  const float alpha = ex[(size_t)e * HEADS + h] / fmaxf(den, 1e-9f);
  float4 v = *(const float4*)(fs + (size_t)s * HD + f);
  float* out = rst + (size_t)d * HD + f;
  atomicAdd(out + 0, alpha * v.x);
  atomicAdd(out + 1, alpha * v.y);
  atomicAdd(out + 2, alpha * v.z);
  atomicAdd(out + 3, alpha * v.w);
}

// rst = (rst + res + bias), optional ReLU (in place)
__global__ void add_res_bias(float* __restrict__ rst,
                             const float* __restrict__ res,
                             const float* __restrict__ b,
                             long total, int Cc, int do_relu) {
  long i = (long)blockIdx.x * blockDim.x + threadIdx.x;
  if (i >= total) return;
  float v = rst[i] + res[i] + b[i % Cc];
  rst[i] = do_relu ? fmaxf(v, 0.f) : v;
}

// Per-channel mean / rsqrt(var+eps) over N rows; one block per channel.
__global__ void bn_stats(const float* __restrict__ v, float* __restrict__ mu,
                         float* __restrict__ rsig, int Cc) {
  const int c = blockIdx.x;
  float s = 0.f, s2 = 0.f;
  for (int r = threadIdx.x; r < NN; r += blockDim.x) {
    float x = v[(size_t)r * Cc + c];
    s += x; s2 += x * x;
  }
  __shared__ float sh[256], sh2[256];
  sh[threadIdx.x] = s; sh2[threadIdx.x] = s2;
  __syncthreads();
  for (int o = 128; o > 0; o >>= 1) {
    if (threadIdx.x < o) { sh[threadIdx.x] += sh[threadIdx.x + o];
                           sh2[threadIdx.x] += sh2[threadIdx.x + o]; }
    __syncthreads();
  }
  if (threadIdx.x == 0) {
    float mean = sh[0] / (float)NN;
    float var  = sh2[0] / (float)NN - mean * mean;  // biased, like torch BN
    mu[c] = mean;
    rsig[c] = rsqrtf(var + BN_EPS);
  }
}

__global__ void bn_apply_relu(const float* __restrict__ v,
                              const float* __restrict__ g,
                              const float* __restrict__ be,
                              const float* __restrict__ mu,
                              const float* __restrict__ rsig,
                              float* __restrict__ out, long total, int Cc) {
  long i = (long)blockIdx.x * blockDim.x + threadIdx.x;
  if (i >= total) return;
  int c = (int)(i % Cc);
  float t = g[c] * (v[i] - mu[c]) * rsig[c] + be[c];
  out[i] = fmaxf(t, 0.f);
}

// Final: mean over heads then log_softmax over CCLS classes.
__global__ void head_mean_logsoftmax(const float* __restrict__ v,
                                     float* __restrict__ out) {
  int n = blockIdx.x * blockDim.x + threadIdx.x;
  if (n >= NN) return;
  const float* row = v + (size_t)n * (HEADS * CCLS);
  float logits[CCLS];
  float mx = -INFINITY;
#pragma unroll
  for (int c = 0; c < CCLS; ++c) {
    float t = 0.25f * (row[c] + row[CCLS + c] + row[2 * CCLS + c] + row[3 * CCLS + c]);
    logits[c] = t;
    mx = fmaxf(mx, t);
  }
  float ssum = 0.f;
#pragma unroll
  for (int c = 0; c < CCLS; ++c) ssum += expf(logits[c] - mx);
  float lse = mx + logf(ssum);
#pragma unroll
  for (int c = 0; c < CCLS; ++c) out[(size_t)n * CCLS + c] = logits[c] - lse;
}

// ---------------------------------------------------------------------------
static inline int cdiv(long a, int b) { return (int)((a + b - 1) / b); }

extern "C" void kernel_launch(void* const* d_in, const int* in_sizes, int n_in,
                              void* d_out, int out_size, void* d_ws, size_t ws_size,
                              hipStream_t stream) {
  const float* x     = (const float*)d_in[0];
  const int*   src   = (const int*)d_in[1];
  const int*   dst   = (const int*)d_in[2];
  const float* Wsrc0 = (const float*)d_in[3];
  const float* Wdst0 = (const float*)d_in[4];
  const float* b0    = (const float*)d_in[5];
  const float* attn0 = (const float*)d_in[6];
  const float* resW0 = (const float*)d_in[7];
  const float* Wsrc1 = (const float*)d_in[8];
  const float* Wdst1 = (const float*)d_in[9];
  const float* b1    = (const float*)d_in[10];
  const float* attn1 = (const float*)d_in[11];
  const float* Wsrc2 = (const float*)d_in[12];
  const float* Wdst2 = (const float*)d_in[13];
  const float* b2    = (const float*)d_in[14];
  const float* attn2 = (const float*)d_in[15];
  const float* resW2 = (const float*)d_in[16];
  const float* g0    = (const float*)d_in[17];
  const float* be0   = (const float*)d_in[18];
  const float* g1    = (const float*)d_in[19];
  const float* be1   = (const float*)d_in[20];
  (void)in_sizes; (void)n_in; (void)out_size; (void)ws_size;

  // ---- workspace carve-up (~130 MB) ----
  const size_t NHD = (size_t)NN * 256;
  float* ws    = (float*)d_ws;
  float* h     = ws;  ws += NHD;                 // layer outputs (max width 256)
  float* fs    = ws;  ws += NHD;                 // fc_src features
  float* fdrst = ws;  ws += NHD;                 // fc_dst, then reused as rst
  float* res   = ws;  ws += NHD;                 // residual projection
  float* score = ws;  ws += (size_t)EE * HEADS;  // per-edge score / exp
  float* mbuf  = ws;  ws += (size_t)NN * HEADS;  // segment max
  float* denom = ws;  ws += (size_t)NN * HEADS;  // segment sum of exp
  float* mu    = ws;  ws += 256;
  float* rsig  = ws;  ws += 256;

  auto gemm = [&](const float* A, const float* B, float* C, int K, int Nout) {
    dim3 grid(NN / (16 * MT), (Nout + 127) / 128);
    gemm_wmma_f32<<<grid, 256, 0, stream>>>(A, B, C, K, Nout);
  };

  const int EH = EE * HEADS;
  auto run_edges = [&](const float* fsb, float* fdb, const float* attn, int Dh) {
    const long NHd = (long)NN * HEADS * Dh;
    fill_f32<<<cdiv((long)NN * HEADS, 256), 256, 0, stream>>>(mbuf, -INFINITY, (long)NN * HEADS);
    edge_score_max<<<cdiv(EH, 256), 256, 0, stream>>>(fsb, fdb, src, dst, attn, score, mbuf, Dh);
    fill_f32<<<cdiv((long)NN * HEADS, 256), 256, 0, stream>>>(denom, 0.f, (long)NN * HEADS);
    fill_f32<<<cdiv(NHd, 256), 256, 0, stream>>>(fdb, 0.f, NHd);  // fd dead -> rst
    edge_exp_denom<<<cdiv(EH, 256), 256, 0, stream>>>(score, mbuf, dst, denom);
    const long tot = (long)EE * (HEADS * Dh / 4);
    edge_aggregate<<<cdiv(tot, 256), 256, 0, stream>>>(score, denom, fsb, src, dst, fdb, Dh);
  };

  // -------- Layer 0: 128 -> 4x64, projected residual, ReLU, BN+ReLU --------
  gemm(x, Wsrc0, fs, IN_DIM, 256);
  gemm(x, Wdst0, fdrst, IN_DIM, 256);
  gemm(x, resW0, res, IN_DIM, 256);
  run_edges(fs, fdrst, attn0, DD);
  add_res_bias<<<cdiv((long)NHD, 256), 256, 0, stream>>>(fdrst, res, b0, (long)NHD, 256, 1);
  bn_stats<<<256, 256, 0, stream>>>(fdrst, mu, rsig, 256);
  bn_apply_relu<<<cdiv((long)NHD, 256), 256, 0, stream>>>(fdrst, g0, be0, mu, rsig, h, (long)NHD, 256);

  // -------- Layer 1: 256 -> 4x64, identity residual, ReLU, BN+ReLU ---------
  gemm(h, Wsrc1, fs, 256, 256);
  gemm(h, Wdst1, fdrst, 256, 256);
  run_edges(fs, fdrst, attn1, DD);
  add_res_bias<<<cdiv((long)NHD, 256), 256, 0, stream>>>(fdrst, h, b1, (long)NHD, 256, 1);
  bn_stats<<<256, 256, 0, stream>>>(fdrst, mu, rsig, 256);
  bn_apply_relu<<<cdiv((long)NHD, 256), 256, 0, stream>>>(fdrst, g1, be1, mu, rsig, h, (long)NHD, 256);

  // -------- Layer 2: 256 -> 4x40, projected residual, no activation --------
  gemm(h, Wsrc2, fs, 256, 160);
  gemm(h, Wdst2, fdrst, 256, 160);
  gemm(h, resW2, res, 256, 160);
  run_edges(fs, fdrst, attn2, CCLS);
  const long tot2 = (long)NN * 160;
  add_res_bias<<<cdiv(tot2, 256), 256, 0, stream>>>(fdrst, res, b2, tot2, 160, 0);
  head_mean_logsoftmax<<<cdiv(NN, 256), 256, 0, stream>>>(fdrst, (float*)d_out);
}